// LocalMatchLoss_72164040507526
// MI455X (gfx1250) — compile-verified
//
#include <hip/hip_runtime.h>
#include <hip/hip_bf16.h>

typedef __attribute__((ext_vector_type(2))) float v2f;
typedef __attribute__((ext_vector_type(8))) float v8f;

#define NB 32
#define NK 8
#define NT 28
#define NHW 64
#define OUTS 5

// D = A(16x4,f32) * B(4x16,f32) + C(16x16,f32); f32-exact pooling contraction.
__device__ __forceinline__ v8f wmma_f32_4(v2f a, v2f b, v8f c) {
  return __builtin_amdgcn_wmma_f32_16x16x4_f32(false, a, false, b, (short)0, c,
                                               false, false);
}

// One workgroup per (b,t): 8 waves, wave w handles class k=w.
__global__ __launch_bounds__(256) void lml_stage1(
    const float* __restrict__ inp,      // [32,8,28,64,64]
    const float* __restrict__ kw,       // [8]
    const int* __restrict__ tgt,        // int64 [32,28,64,64] viewed as int pairs
    float* __restrict__ partials) {     // [896]
  __shared__ float scnt[NK][OUTS][OUTS];   // one-hot window counts
  __shared__ float Yt[NK][16][64];         // Y transposed: Yt[k][n][row] = Y[row][n]
  __shared__ float sbuf[NK];

  const int bt = blockIdx.x;            // 0..895
  const int tid = threadIdx.x;
  const int lane = tid & 31;
  const int wave = tid >> 5;            // == k
  const int nl = lane & 15;             // N (or M) index within tile
  const int hi = lane >> 4;             // lane-half selects K pair {0,1} vs {2,3}

  if (tid < NK * OUTS * OUTS) ((float*)scnt)[tid] = 0.0f;
  __syncthreads();

  // ---- Phase A: window histogram of target==k (read target once per WG) ----
  const int* tg = tgt + (size_t)bt * (NHW * NHW) * 2;  // int64 -> low word
  #pragma unroll
  for (int it = 0; it < 16; ++it) {
    int p = it * 256 + tid;
    int h = p >> 6, w = p & 63;
    int kv = tg[2 * p];                 // 0..7
    int r0 = h < 13 ? 0 : h < 26 ? 1 : h < 39 ? 2 : h < 52 ? 3 : 4;
    int c0 = w < 13 ? 0 : w < 26 ? 1 : w < 39 ? 2 : w < 52 ? 3 : 4;
    bool rx = (h == 12) || (h == 25) || (h == 38) || (h == 51);  // overlap rows
    bool cx = (w == 12) || (w == 25) || (w == 38) || (w == 51);  // overlap cols
    atomicAdd(&scnt[kv][r0][c0], 1.0f);
    if (rx) atomicAdd(&scnt[kv][r0 + 1][c0], 1.0f);
    if (cx) atomicAdd(&scnt[kv][r0][c0 + 1], 1.0f);
    if (rx && cx) atomicAdd(&scnt[kv][r0 + 1][c0 + 1], 1.0f);
  }
  __syncthreads();

  // Per-lane window bounds for index nl: ws=floor(64n/5), we=ceil(64(n+1)/5).
  // Branchless membership: memb(c) = (unsigned)(c - wlo) < rng; rng=0 kills n>=5.
  const int wlo = (nl * 64) / 5;
  const unsigned rng = (nl < OUTS) ? (unsigned)((nl * 64 + 68) / 5 - wlo) : 0u;

  // Indicator operands are identical for product-1 B (Pw) and product-2 A (Ph^T),
  // and invariant across the r-loop: compute the 16 v2f pairs once, reuse 5x.
  v2f pm[16];
  #pragma unroll
  for (int q = 0; q < 16; ++q) {
    int c = (q << 2) + (hi << 1);       // K columns {c, c+1} for this lane-half
    pm[q].x = ((unsigned)(c - wlo) < rng) ? 1.0f : 0.0f;
    pm[q].y = ((unsigned)(c + 1 - wlo) < rng) ? 1.0f : 0.0f;
  }

  // ---- Phase B: Y(64x16) = X(64x64) * Pw(64x16) via WMMA ----
  const int bb = bt / NT, tt = bt % NT;
  const float* Xp = inp + (((size_t)bb * NK + wave) * NT + tt) * (NHW * NHW);
  float* yt = &Yt[wave][0][0];          // yt[n*64 + row]

  #pragma unroll
  for (int r = 0; r < 4; ++r) {
    v8f acc = {};
    const float* xr = Xp + ((r << 4) + nl) * 64;  // A row = 16r + (lane&15)
    #pragma unroll
    for (int q = 0; q < 16; ++q) {
      int c = (q << 2) + (hi << 1);
      v2f a = *(const v2f*)(xr + c);    // global_load_b64, each element read once
      acc = wmma_f32_4(a, pm[q], acc);
    }
    // D layout -> Yt: lane half selects M=v (rows 16r+v) or M=v+8
    int rb = (r << 4) + (hi << 3);
    #pragma unroll
    for (int v = 0; v < 8; ++v) yt[nl * 64 + rb + v] = acc[v];
  }
  __syncthreads();

  // ---- Phase C: Z(16x16) = Ph^T(16x64) * Y(64x16) via WMMA ----
  v8f z = {};
  #pragma unroll
  for (int p = 0; p < 16; ++p) {
    int c = (p << 2) + (hi << 1);       // K rows {c, c+1}
    v2f bm = *(const v2f*)(yt + nl * 64 + c);  // Y[c][nl], Y[c+1][nl] (ds_load_b64)
    z = wmma_f32_4(pm[p], bm, z);       // A[i][kk] = Ph[c][i], i = nl
  }

  // ---- Phase D: |in_sum - cnt| / area, reduce ----
  float partial = 0.0f;
  if (lane < OUTS) {                    // Z[i][j]: vgpr i (<5), lane j (<5)
    int j = lane;
    float rw = (float)rng;              // we[j]-ws[j]
    #pragma unroll
    for (int i = 0; i < OUTS; ++i) {
      float rh = (float)((i * 64 + 68) / 5 - (i * 64) / 5);
      partial += __builtin_fabsf(z[i] - scnt[wave][i][j]) / (rh * rw);
    }
  }
  #pragma unroll
  for (int off = 16; off > 0; off >>= 1)
    partial += __shfl_down(partial, off, 32);
  if (lane == 0) sbuf[wave] = partial * kw[wave];
  __syncthreads();
  if (tid == 0) {
    float s = 0.0f;
    #pragma unroll
    for (int k = 0; k < NK; ++k) s += sbuf[k];
    partials[bt] = s;                   // fully overwritten every call
  }
}

// Deterministic final reduction of 896 partials; applies mean factor 1/(32*25).
__global__ __launch_bounds__(256) void lml_stage2(const float* __restrict__ ws,
                                                  float* __restrict__ out) {
  __shared__ float s[256];
  float v = 0.0f;
  for (int i = threadIdx.x; i < NB * NT; i += 256) v += ws[i];
  s[threadIdx.x] = v;
  __syncthreads();
  for (int o = 128; o > 0; o >>= 1) {
    if (threadIdx.x < o) s[threadIdx.x] += s[threadIdx.x + o];
    __syncthreads();
  }
  if (threadIdx.x == 0) out[0] = s[0] * (1.0f / (float)(NB * OUTS * OUTS));
}

extern "C" void kernel_launch(void* const* d_in, const int* in_sizes, int n_in,
                              void* d_out, int out_size, void* d_ws, size_t ws_size,
                              hipStream_t stream) {
  const float* inp = (const float*)d_in[0];   // [32,8,28,64,64] f32
  const float* kw  = (const float*)d_in[1];   // [8] f32
  const int*   tgt = (const int*)d_in[2];     // [32,28,64,64] int64
  float* out = (float*)d_out;                 // scalar f32
  float* ws  = (float*)d_ws;                  // >= 896 floats

  lml_stage1<<<NB * NT, 256, 0, stream>>>(inp, kw, tgt, ws);
  lml_stage2<<<1, 256, 0, stream>>>(ws, out);
}